// SPDNet_18373870092355
// MI455X (gfx1250) — compile-verified
//
#include <hip/hip_runtime.h>

// ---------------------------------------------------------------------------
// SPDNet forward, fused for MI455X (gfx1250, wave32).
//   Stage 0 (TDM):  tensor_load_to_lds stages W (60x20) as a zero-padded
//                   60x32 LDS tile (TDM OOB-read-returns-zero => free padding)
//   Stage 1 (WMMA f32 16x16x4): y = W^T X W   (60x60 -> 20x20, per matrix)
//   Stage 2 (fused): ONE symmetric eigendecomposition (cyclic Jacobi in LDS);
//                   eigenvalue map f(e) = log(max(e, EPS)) realizes BOTH the
//                   SPDRectified and the tangent-space log (exact fusion:
//                   U diag(max(e,EPS)) U^T is already an eigendecomposition).
//   Stage 3: triu vectorization with sqrt(2) off-diag -> feat[4096][840] (ws)
//   Kernel 2: out = feat @ lin_w^T + lin_b  (coalesced wave-per-row + shuffle)
//
// Roofline: X traffic 16384*14.4KB = 236MB ~ 10us @ 23.3TB/s; bilinear is
// HBM-bound with f32 WMMA, so no precision drop is warranted. Jacobi phase is
// the VALU/LDS hotspot -> keep 4 waves/block for occupancy.
// ---------------------------------------------------------------------------

typedef __attribute__((ext_vector_type(2))) float v2f;
typedef __attribute__((ext_vector_type(8))) float v8f;
typedef __attribute__((ext_vector_type(4))) unsigned int u32x4;
typedef __attribute__((ext_vector_type(8))) unsigned int u32x8;

#define N_DIM 60
#define M_DIM 20
#define VECN  210
#define EPS_F 1e-4f
#define WAVES 4            // waves (= matrices) per block
#define SCR_F 2112         // per-wave LDS scratch, floats (Z pad 32x65=2080)
#define JAC_SWEEPS 8
#define WPAD  32           // padded W row stride (cols 20..31 are zero)

// Assumed gfx1250 WMMA f32 16x16x4 fragment layouts (ISA 7.12.2):
//  A (16x4):  lanes 0-15: {A[l][k0], A[l][k0+1]}   lanes 16-31: {A[l-16][k0+2], A[l-16][k0+3]}
//  B (4x16):  lanes 0-15: {B[k0][l], B[k0+1][l]}   lanes 16-31: {B[k0+2][l-16], B[k0+3][l-16]}
//  C/D:       VGPR v: lanes 0-15 -> D[v][l], lanes 16-31 -> D[v+8][l-16]

__global__ __launch_bounds__(WAVES * 32) void spd_fused_kernel(
    const float* __restrict__ X,      // [nmat][60][60]
    const float* __restrict__ W,      // [60][20]
    float* __restrict__ feat,         // [nmat][210]
    int nmat)
{
    __shared__ float sW[N_DIM * WPAD];               // 7.7 KB zero-padded W tile
    __shared__ float scr_all[WAVES][SCR_F];          // 33.8 KB per-wave scratch

    const int tid  = threadIdx.x;
    const int wave = tid >> 5;
    const int lane = tid & 31;
    const int hh   = lane >> 4;      // half-wave selector (k pair)
    const int lid  = lane & 15;

    // ---- Stage 0: TDM load of W into LDS as a padded 60x32 tile -----------
    // D# group0: count=1 | lds_addr | global_addr | type=2("image")
    // D# group1: data_size=4B, tensor_dim0=20, tensor_dim1=60,
    //            tile_dim0=32 (OOB cols 20..31 read as zero), tile_dim1=60,
    //            tensor_dim0_stride=20.  Groups 2/3 omitted (2-D tensor).
    if (wave == 0) {
        const unsigned long long wga = (unsigned long long)(const void*)W;
        const unsigned lds_off = (unsigned)(unsigned long long)(void*)&sW[0];
        u32x4 g0;
        g0.x = 1u;                                   // count = 1 valid D#
        g0.y = lds_off;                              // lds_addr (bytes)
        g0.z = (unsigned)(wga & 0xffffffffu);        // global_addr[31:0]
        g0.w = (unsigned)((wga >> 32) & 0x01ffffffu) // global_addr[56:32]
             | (2u << 30);                           // type = 2
        u32x8 g1;
        g1[0] = (2u << 16);                          // data_size = 4 bytes
        g1[1] = (unsigned)M_DIM << 16;               // tensor_dim0 = 20
        g1[2] = (unsigned)N_DIM << 16;               // tensor_dim1 = 60
        g1[3] = (unsigned)WPAD << 16;                // tile_dim0   = 32
        g1[4] = (unsigned)N_DIM;                     // tile_dim1   = 60
        g1[5] = (unsigned)M_DIM;                     // tensor_dim0_stride = 20
        g1[6] = 0u;
        g1[7] = 0u;
        asm volatile("tensor_load_to_lds %0, %1" :: "s"(g0), "s"(g1) : "memory");
        __builtin_amdgcn_s_wait_tensorcnt(0);
    }
    __syncthreads();                                 // publish sW to all waves

    const int idx = blockIdx.x * WAVES + wave;
    if (idx >= nmat) return;

    float* scr = scr_all[wave];
    const float* Xm = X + (size_t)idx * (N_DIM * N_DIM);

    // Prefetch this matrix's X (128B per lane-step => whole 14.4KB tile)
    for (int o = lane * 32; o < N_DIM * N_DIM; o += 32 * 32)
        __builtin_prefetch(&Xm[o], 0, 3);

    // ---------------- Stage A: Z = W^T X  (padded 32 x 64) ----------------
    // Z stored [32][65] in LDS (pad 65 -> conflict-free column gathers later)
    float* Z = scr;
    for (int mt = 0; mt < 2; ++mt) {
        v8f acc[4] = {v8f{}, v8f{}, v8f{}, v8f{}};
        const int m = mt * 16 + lid;                 // 0..31, sW zero-padded
        #pragma unroll
        for (int k = 0; k < 15; ++k) {               // K = 60 -> 15 steps of 4
            const int n0 = 4 * k + 2 * hh;
            v2f a;                                   // W^T[m][n] = W[n][m]
            a.x = sW[(n0 + 0) * WPAD + m];
            a.y = sW[(n0 + 1) * WPAD + m];
            #pragma unroll
            for (int pt = 0; pt < 4; ++pt) {
                const int p  = pt * 16 + lid;
                const int pc = (p < N_DIM) ? p : (N_DIM - 1);  // clamp: load
                const float bx = Xm[(n0 + 0) * N_DIM + pc];    // is always
                const float by = Xm[(n0 + 1) * N_DIM + pc];    // in-bounds
                v2f b;
                b.x = (p < N_DIM) ? bx : 0.f;        // branch-free v_cndmask
                b.y = (p < N_DIM) ? by : 0.f;
                acc[pt] = __builtin_amdgcn_wmma_f32_16x16x4_f32(
                    false, a, false, b, (short)0, acc[pt], false, false);
            }
        }
        #pragma unroll
        for (int pt = 0; pt < 4; ++pt)
            #pragma unroll
            for (int v = 0; v < 8; ++v)
                Z[(mt * 16 + v + 8 * hh) * 65 + pt * 16 + lid] = acc[pt][v];
    }

    // ---------------- Stage B: y = Z W  (20x20, padded 32x32) --------------
    v8f yacc[4] = {v8f{}, v8f{}, v8f{}, v8f{}};      // [mt*2+qt]
    #pragma unroll
    for (int k = 0; k < 15; ++k) {
        const int n0 = 4 * k + 2 * hh;
        v2f a0, a1, b0, b1;
        a0.x = Z[(lid)      * 65 + n0];  a0.y = Z[(lid)      * 65 + n0 + 1];
        a1.x = Z[(16 + lid) * 65 + n0];  a1.y = Z[(16 + lid) * 65 + n0 + 1];
        b0.x = sW[(n0 + 0) * WPAD + lid];            // q = 0..15
        b0.y = sW[(n0 + 1) * WPAD + lid];
        b1.x = sW[(n0 + 0) * WPAD + 16 + lid];       // q = 16..31 (>=20 zero)
        b1.y = sW[(n0 + 1) * WPAD + 16 + lid];
        yacc[0] = __builtin_amdgcn_wmma_f32_16x16x4_f32(false, a0, false, b0, (short)0, yacc[0], false, false);
        yacc[1] = __builtin_amdgcn_wmma_f32_16x16x4_f32(false, a0, false, b1, (short)0, yacc[1], false, false);
        yacc[2] = __builtin_amdgcn_wmma_f32_16x16x4_f32(false, a1, false, b0, (short)0, yacc[2], false, false);
        yacc[3] = __builtin_amdgcn_wmma_f32_16x16x4_f32(false, a1, false, b1, (short)0, yacc[3], false, false);
    }

    // ---------------- Stage C: Jacobi eigensolver in LDS -------------------
    // Alias over Z (Z is dead). A[20][21], V[20][21], rotation buffers.
    float* Am  = scr;            // 0   .. 419
    float* Vm  = scr + 420;      // 420 .. 839
    float* cb  = scr + 840;      // 10 rotations: cos
    float* sb  = scr + 852;      // sin
    float* pfb = scr + 864;      // p (as float)
    float* qfb = scr + 876;      // q (as float)
    float* eb  = scr + 888;      // 20 log-eigenvalues

    #pragma unroll
    for (int mt = 0; mt < 2; ++mt)
        #pragma unroll
        for (int qt = 0; qt < 2; ++qt)
            #pragma unroll
            for (int v = 0; v < 8; ++v) {
                const int mm = mt * 16 + v + 8 * hh;
                const int qq = qt * 16 + lid;
                if (mm < M_DIM && qq < M_DIM)
                    Am[mm * 21 + qq] = yacc[mt * 2 + qt][v];
            }
    for (int t = lane; t < 420; t += 32) {
        const int i = t / 21, j = t % 21;
        Vm[t] = (i == j) ? 1.f : 0.f;
    }

    for (int sweep = 0; sweep < JAC_SWEEPS; ++sweep) {
        for (int r = 0; r < 19; ++r) {
            // Tournament (circle) ordering: 10 disjoint pairs per round,
            // covering all 190 index pairs over 19 rounds.
            if (lane < 10) {
                int p, q;
                if (lane == 0) { p = 0; q = 1 + ((18 + r) % 19); }
                else {
                    p = 1 + ((lane - 1 + r) % 19);
                    q = 1 + ((18 - lane + r) % 19);
                }
                if (p > q) { int t2 = p; p = q; q = t2; }
                const float app = Am[p * 21 + p];
                const float aqq = Am[q * 21 + q];
                const float apq = Am[p * 21 + q];
                float c = 1.f, s = 0.f;
                if (__builtin_fabsf(apq) > 1e-20f) {
                    const float tau = (aqq - app) / (2.f * apq);
                    float t = 1.f / (__builtin_fabsf(tau) + __builtin_sqrtf(1.f + tau * tau));
                    if (tau < 0.f) t = -t;
                    c = 1.f / __builtin_sqrtf(1.f + t * t);
                    s = t * c;
                }
                cb[lane] = c; sb[lane] = s;
                pfb[lane] = (float)p; qfb[lane] = (float)q;
            }
            // Column update: A <- A J, V <- V J  (200 disjoint (i,pair) items)
            for (int it = lane; it < 200; it += 32) {
                const int tt = it / 20, i = it % 20;
                const int p = (int)pfb[tt], q = (int)qfb[tt];
                const float c = cb[tt], s = sb[tt];
                const float aip = Am[i * 21 + p], aiq = Am[i * 21 + q];
                Am[i * 21 + p] = c * aip - s * aiq;
                Am[i * 21 + q] = s * aip + c * aiq;
                const float vip = Vm[i * 21 + p], viq = Vm[i * 21 + q];
                Vm[i * 21 + p] = c * vip - s * viq;
                Vm[i * 21 + q] = s * vip + c * viq;
            }
            // Row update: A <- J^T A
            for (int it = lane; it < 200; it += 32) {
                const int tt = it / 20, i = it % 20;
                const int p = (int)pfb[tt], q = (int)qfb[tt];
                const float c = cb[tt], s = sb[tt];
                const float api = Am[p * 21 + i], aqi = Am[q * 21 + i];
                Am[p * 21 + i] = c * api - s * aqi;
                Am[q * 21 + i] = s * api + c * aqi;
            }
        }
    }

    // Fused rectify+log eigenvalue map
    for (int k = lane; k < M_DIM; k += 32)
        eb[k] = __builtin_logf(fmaxf(Am[k * 21 + k], EPS_F));

    // -------- Stage D: L = V diag(log e) V^T, triu-vectorize, store --------
    const float SQRT2 = 1.41421356237309504880f;
    float* out = feat + (size_t)idx * VECN;
    for (int vi = lane; vi < VECN; vi += 32) {
        int i = 0, rem = vi;                         // row-major triu index
        while (rem >= M_DIM - i) { rem -= M_DIM - i; ++i; }
        const int j = i + rem;
        float acc = 0.f;
        #pragma unroll
        for (int k = 0; k < M_DIM; ++k)
            acc = fmaf(Vm[i * 21 + k] * eb[k], Vm[j * 21 + k], acc);
        out[vi] = (i == j) ? acc : SQRT2 * acc;
    }
}

// ---------------------------------------------------------------------------
// Kernel 2: out[rows][11] = feat[rows][840] @ lin_w[11][840]^T + lin_b
// One wave per row: lanes stride K (fully coalesced feat reads), shuffle-reduce.
// ---------------------------------------------------------------------------
__global__ __launch_bounds__(256) void spd_linear_kernel(
    const float* __restrict__ feat, const float* __restrict__ lw,
    const float* __restrict__ lb, float* __restrict__ out, int rows)
{
    const int gw   = (blockIdx.x * blockDim.x + threadIdx.x) >> 5;
    const int lane = threadIdx.x & 31;
    if (gw >= rows) return;
    const float* f = feat + (size_t)gw * 840;
    float acc[11];
    #pragma unroll
    for (int j = 0; j < 11; ++j) acc[j] = 0.f;
    for (int k = lane; k < 840; k += 32) {
        const float fv = f[k];
        #pragma unroll
        for (int j = 0; j < 11; ++j)
            acc[j] = fmaf(fv, lw[j * 840 + k], acc[j]);
    }
    #pragma unroll
    for (int j = 0; j < 11; ++j) {
        float v = acc[j];
        #pragma unroll
        for (int off = 16; off > 0; off >>= 1)
            v += __shfl_down(v, off, 32);
        if (lane == 0) out[(size_t)gw * 11 + j] = v + lb[j];
    }
}

extern "C" void kernel_launch(void* const* d_in, const int* in_sizes, int n_in,
                              void* d_out, int out_size, void* d_ws, size_t ws_size,
                              hipStream_t stream)
{
    const float* x  = (const float*)d_in[0];   // [4096,4,60,60] f32
    const float* W  = (const float*)d_in[1];   // [60,20] f32
    const float* lw = (const float*)d_in[2];   // [11,840] f32
    const float* lb = (const float*)d_in[3];   // [11] f32
    float* out  = (float*)d_out;               // [4096,11] f32
    float* feat = (float*)d_ws;                // [16384,210] f32 = 13.8 MB

    const int nmat = in_sizes[0] / (N_DIM * N_DIM);   // 16384
    const int rows = out_size / 11;                   // 4096

    dim3 b1(WAVES * 32);
    dim3 g1((nmat + WAVES - 1) / WAVES);
    spd_fused_kernel<<<g1, b1, 0, stream>>>(x, W, feat, nmat);

    dim3 b2(256);
    dim3 g2((rows + 7) / 8);                          // 8 waves per block
    spd_linear_kernel<<<g2, b2, 0, stream>>>(feat, lw, lb, out, rows);
}